// WeightedClassLoss_46196668236574
// MI455X (gfx1250) — compile-verified
//
#include <hip/hip_runtime.h>
#include <hip/hip_bf16.h>
#include <math.h>
#include <stdint.h>

// ---------------------------------------------------------------------------
// Weighted quantile-class MSE loss for MI455X (gfx1250).
// Exact quantiles via 3-level radix select (12+10+10 bits of an order-
// preserving u32 key), then a fused weighted-MSE reduction pass.
// ---------------------------------------------------------------------------

#define TPB 256
#define GRID_STREAM 2048
#define H1_BITS 12
#define H1_BINS (1 << H1_BITS) /* 4096 */
#define L2_BITS 10
#define L2_BINS (1 << L2_BITS) /* 1024 */
#define NRANKS 10

// workspace layout (bytes)
#define OFF_HIST1 0u
#define OFF_HIST2 (OFF_HIST1 + H1_BINS * 4u)
#define OFF_HIST3 (OFF_HIST2 + NRANKS * L2_BINS * 4u)
#define OFF_SEL1B (OFF_HIST3 + NRANKS * L2_BINS * 4u)
#define OFF_SEL1R (OFF_SEL1B + NRANKS * 4u)
#define OFF_SEL2B (OFF_SEL1R + NRANKS * 4u)
#define OFF_SEL2R (OFF_SEL2B + NRANKS * 4u)
#define OFF_VALS (OFF_SEL2R + NRANKS * 4u)
#define OFF_QUANT (OFF_VALS + NRANKS * 4u)
#define OFF_PART (OFF_QUANT + 8u * 4u)
#define ZERO_BYTES OFF_PART

struct RankInfo {
  unsigned ranks[NRANKS];
  float frac[4];
};

#if __has_builtin(__builtin_amdgcn_global_load_async_to_lds_b128) && \
    __has_builtin(__builtin_amdgcn_s_wait_asynccnt)
#define WL_HAVE_ASYNC 1
#else
#define WL_HAVE_ASYNC 0
#endif

#if __has_builtin(__builtin_amdgcn_wmma_f32_16x16x4_f32)
#define WL_HAVE_WMMA 1
#else
#define WL_HAVE_WMMA 0
#endif

typedef __attribute__((ext_vector_type(2))) float wl_v2f;
typedef __attribute__((ext_vector_type(8))) float wl_v8f;
typedef __attribute__((ext_vector_type(4))) int wl_v4i;

// order-preserving float->u32 key and inverse
__device__ __forceinline__ unsigned wl_key(float f) {
  unsigned u = __float_as_uint(f);
  return (u & 0x80000000u) ? ~u : (u | 0x80000000u);
}
__device__ __forceinline__ float wl_unkey(unsigned k) {
  unsigned u = (k & 0x80000000u) ? (k & 0x7fffffffu) : ~k;
  return __uint_as_float(u);
}

// ---------------------------------------------------------------------------
// Pass 1: 4096-bin histogram of key[31:20], LDS-staged.
// On gfx1250 we stream targets through LDS with double-buffered
// GLOBAL_LOAD_ASYNC_TO_LDS_B128 + s_wait_asynccnt.
// ---------------------------------------------------------------------------
__global__ void __launch_bounds__(TPB) wl_hist1(const float* __restrict__ targ,
                                                unsigned* __restrict__ hist1,
                                                unsigned n) {
  __shared__ unsigned h[H1_BINS];
#if WL_HAVE_ASYNC
  __shared__ float4 stage[2][TPB];
#endif
  for (unsigned i = threadIdx.x; i < H1_BINS; i += TPB) h[i] = 0u;
  __syncthreads();

  const unsigned stride = gridDim.x * TPB;
  const unsigned gid = blockIdx.x * TPB + threadIdx.x;
  const unsigned n4 = n >> 2;
  const float4* tv = (const float4*)targ;

#if WL_HAVE_ASYNC
  {
    unsigned idx = gid;
    int buf = 0;
    const bool any = idx < n4;
    auto* lds0 =
        (__attribute__((address_space(3))) wl_v4i*)&stage[0][threadIdx.x];
    auto* lds1 =
        (__attribute__((address_space(3))) wl_v4i*)&stage[1][threadIdx.x];
    if (any)
      __builtin_amdgcn_global_load_async_to_lds_b128((wl_v4i*)(tv + idx), lds0,
                                                     0, 0);
    while (idx + stride < n4) {
      __builtin_amdgcn_global_load_async_to_lds_b128(
          (wl_v4i*)(tv + idx + stride), buf ? lds0 : lds1, 0, 0);
      __builtin_amdgcn_s_wait_asynccnt(1);
      asm volatile("" ::: "memory");
      float4 v = stage[buf][threadIdx.x];
      atomicAdd(&h[wl_key(v.x) >> 20], 1u);
      atomicAdd(&h[wl_key(v.y) >> 20], 1u);
      atomicAdd(&h[wl_key(v.z) >> 20], 1u);
      atomicAdd(&h[wl_key(v.w) >> 20], 1u);
      buf ^= 1;
      idx += stride;
    }
    __builtin_amdgcn_s_wait_asynccnt(0);
    asm volatile("" ::: "memory");
    if (any) {
      float4 v = stage[buf][threadIdx.x];
      atomicAdd(&h[wl_key(v.x) >> 20], 1u);
      atomicAdd(&h[wl_key(v.y) >> 20], 1u);
      atomicAdd(&h[wl_key(v.z) >> 20], 1u);
      atomicAdd(&h[wl_key(v.w) >> 20], 1u);
    }
  }
#else
  for (unsigned i = gid; i < n4; i += stride) {
    __builtin_prefetch((const void*)(tv + i + 8u * stride), 0, 0);
    float4 v = tv[i];
    atomicAdd(&h[wl_key(v.x) >> 20], 1u);
    atomicAdd(&h[wl_key(v.y) >> 20], 1u);
    atomicAdd(&h[wl_key(v.z) >> 20], 1u);
    atomicAdd(&h[wl_key(v.w) >> 20], 1u);
  }
#endif
  // scalar tail
  for (unsigned i = (n4 << 2) + gid; i < n; i += stride)
    atomicAdd(&h[wl_key(targ[i]) >> 20], 1u);

  __syncthreads();
  for (unsigned i = threadIdx.x; i < H1_BINS; i += TPB) {
    unsigned c = h[i];
    if (c) atomicAdd(&hist1[i], c);
  }
}

// ---------------------------------------------------------------------------
// Select level 1: locate the level-1 bin + residual rank for each target rank.
// ---------------------------------------------------------------------------
__global__ void wl_sel1(const unsigned* __restrict__ hist1,
                        unsigned* __restrict__ selb, unsigned* __restrict__ selr,
                        RankInfo ri) {
  if (threadIdx.x == 0 && blockIdx.x == 0) {
    unsigned long long cum = 0;
    int r = 0;
    for (int b = 0; b < H1_BINS && r < NRANKS; ++b) {
      unsigned c = hist1[b];
      while (r < NRANKS && (unsigned long long)ri.ranks[r] < cum + c) {
        selb[r] = (unsigned)b;
        selr[r] = (unsigned)((unsigned long long)ri.ranks[r] - cum);
        ++r;
      }
      cum += c;
    }
  }
}

// ---------------------------------------------------------------------------
// Pass 2: for each rank, 1024-bin histogram of key[19:10] restricted to its
// level-1 bin. 10*1024 u32 = 40KB LDS (WGP has 320KB).
// ---------------------------------------------------------------------------
__global__ void __launch_bounds__(TPB) wl_hist2(const float* __restrict__ targ,
                                                const unsigned* __restrict__ sel1b,
                                                unsigned* __restrict__ hist2,
                                                unsigned n) {
  __shared__ unsigned h[NRANKS * L2_BINS];
  for (unsigned i = threadIdx.x; i < NRANKS * L2_BINS; i += TPB) h[i] = 0u;
  __syncthreads();

  unsigned b1[NRANKS];
#pragma unroll
  for (int r = 0; r < NRANKS; ++r) b1[r] = sel1b[r];

  const unsigned stride = gridDim.x * TPB;
  const unsigned gid = blockIdx.x * TPB + threadIdx.x;
  const unsigned n4 = n >> 2;
  const float4* tv = (const float4*)targ;

  for (unsigned i = gid; i < n4; i += stride) {
    __builtin_prefetch((const void*)(tv + i + 8u * stride), 0, 0);
    float4 v = tv[i];
    float e[4] = {v.x, v.y, v.z, v.w};
#pragma unroll
    for (int c = 0; c < 4; ++c) {
      unsigned key = wl_key(e[c]);
      unsigned hi = key >> 20;
      unsigned mid = (key >> 10) & (L2_BINS - 1);
#pragma unroll
      for (int r = 0; r < NRANKS; ++r)
        if (hi == b1[r]) atomicAdd(&h[r * L2_BINS + mid], 1u);
    }
  }
  for (unsigned i = (n4 << 2) + gid; i < n; i += stride) {
    unsigned key = wl_key(targ[i]);
    unsigned hi = key >> 20;
    unsigned mid = (key >> 10) & (L2_BINS - 1);
#pragma unroll
    for (int r = 0; r < NRANKS; ++r)
      if (hi == b1[r]) atomicAdd(&h[r * L2_BINS + mid], 1u);
  }
  __syncthreads();
  for (unsigned i = threadIdx.x; i < NRANKS * L2_BINS; i += TPB) {
    unsigned c = h[i];
    if (c) atomicAdd(&hist2[i], c);
  }
}

__global__ void wl_sel2(const unsigned* __restrict__ hist2,
                        const unsigned* __restrict__ sel1r,
                        unsigned* __restrict__ sel2b, unsigned* __restrict__ sel2r) {
  int r = (int)threadIdx.x;
  if (r < NRANKS && blockIdx.x == 0) {
    unsigned rank = sel1r[r], cum = 0;
    for (int b = 0; b < L2_BINS; ++b) {
      unsigned c = hist2[r * L2_BINS + b];
      if (rank < cum + c) {
        sel2b[r] = (unsigned)b;
        sel2r[r] = rank - cum;
        break;
      }
      cum += c;
    }
  }
}

// ---------------------------------------------------------------------------
// Pass 3: final 10 bits key[9:0] restricted to (b1,b2).
// ---------------------------------------------------------------------------
__global__ void __launch_bounds__(TPB) wl_hist3(const float* __restrict__ targ,
                                                const unsigned* __restrict__ sel1b,
                                                const unsigned* __restrict__ sel2b,
                                                unsigned* __restrict__ hist3,
                                                unsigned n) {
  __shared__ unsigned h[NRANKS * L2_BINS];
  for (unsigned i = threadIdx.x; i < NRANKS * L2_BINS; i += TPB) h[i] = 0u;
  __syncthreads();

  unsigned hb[NRANKS];  // combined high 22 bits
#pragma unroll
  for (int r = 0; r < NRANKS; ++r) hb[r] = (sel1b[r] << 10) | sel2b[r];

  const unsigned stride = gridDim.x * TPB;
  const unsigned gid = blockIdx.x * TPB + threadIdx.x;
  const unsigned n4 = n >> 2;
  const float4* tv = (const float4*)targ;

  for (unsigned i = gid; i < n4; i += stride) {
    __builtin_prefetch((const void*)(tv + i + 8u * stride), 0, 0);
    float4 v = tv[i];
    float e[4] = {v.x, v.y, v.z, v.w};
#pragma unroll
    for (int c = 0; c < 4; ++c) {
      unsigned key = wl_key(e[c]);
      unsigned top = key >> 10;
      unsigned lo = key & (L2_BINS - 1);
#pragma unroll
      for (int r = 0; r < NRANKS; ++r)
        if (top == hb[r]) atomicAdd(&h[r * L2_BINS + lo], 1u);
    }
  }
  for (unsigned i = (n4 << 2) + gid; i < n; i += stride) {
    unsigned key = wl_key(targ[i]);
    unsigned top = key >> 10;
    unsigned lo = key & (L2_BINS - 1);
#pragma unroll
    for (int r = 0; r < NRANKS; ++r)
      if (top == hb[r]) atomicAdd(&h[r * L2_BINS + lo], 1u);
  }
  __syncthreads();
  for (unsigned i = threadIdx.x; i < NRANKS * L2_BINS; i += TPB) {
    unsigned c = h[i];
    if (c) atomicAdd(&hist3[i], c);
  }
}

// Resolve final values and build the 6 quantiles (f32 interpolation like jnp).
__global__ void wl_sel3(const unsigned* __restrict__ hist3,
                        const unsigned* __restrict__ sel1b,
                        const unsigned* __restrict__ sel2b,
                        const unsigned* __restrict__ sel2r,
                        float* __restrict__ vals, float* __restrict__ quant,
                        RankInfo ri) {
  __shared__ float v[NRANKS];
  int r = (int)threadIdx.x;
  if (r < NRANKS && blockIdx.x == 0) {
    unsigned rank = sel2r[r], cum = 0, b3 = 0;
    for (int b = 0; b < L2_BINS; ++b) {
      unsigned c = hist3[r * L2_BINS + b];
      if (rank < cum + c) {
        b3 = (unsigned)b;
        break;
      }
      cum += c;
    }
    unsigned key = (sel1b[r] << 20) | (sel2b[r] << 10) | b3;
    v[r] = wl_unkey(key);
    vals[r] = v[r];
  }
  __syncthreads();
  if (threadIdx.x == 0 && blockIdx.x == 0) {
    quant[0] = v[0];
    quant[5] = v[9];
#pragma unroll
    for (int j = 1; j <= 4; ++j) {
      float a = v[2 * j - 1], b = v[2 * j];
      quant[j] = a + ri.frac[j - 1] * (b - a);
    }
  }
}

// ---------------------------------------------------------------------------
// Fused weighted-MSE pass. Deterministic reduction:
// per-thread double -> per-block tree (last 32->16 step via V_WMMA_F32_16X16X4_F32
// ones-matrix trick) -> per-block partial -> one-block final reduce.
// ---------------------------------------------------------------------------
__device__ __forceinline__ float wl_term(float p, float t, float q0, float q1,
                                         float q2, float q3, float q4, float q5) {
  int c = (int)(t > q0) + (int)(t > q1) + (int)(t > q2) + (int)(t > q3) +
          (int)(t > q4) + (int)(t > q5);  // searchsorted(..., left)
  if (c < 1) c = 1;                        // labels[labels==0] = 1
  int d = c - 4;                           // |label-1 - 3|
  d = d < 0 ? -d : d;
  float w = (d == 0) ? 0.0f
            : (d == 1) ? (1.0f / 3.0f)
            : (d == 2) ? (2.0f / 3.0f)
                       : 1.0f;
  float e = p - t;
  return w * e * e;
}

__global__ void __launch_bounds__(TPB) wl_mse(const float* __restrict__ pred,
                                              const float* __restrict__ targ,
                                              const float* __restrict__ quant,
                                              float* __restrict__ partials,
                                              unsigned n) {
  const float q0 = quant[0], q1 = quant[1], q2 = quant[2];
  const float q3 = quant[3], q4 = quant[4], q5 = quant[5];
  const unsigned stride = gridDim.x * TPB;
  const unsigned gid = blockIdx.x * TPB + threadIdx.x;
  const unsigned n4 = n >> 2;
  const float4* pv = (const float4*)pred;
  const float4* tv = (const float4*)targ;

  double acc = 0.0;
  for (unsigned i = gid; i < n4; i += stride) {
    __builtin_prefetch((const void*)(pv + i + 8u * stride), 0, 0);
    __builtin_prefetch((const void*)(tv + i + 8u * stride), 0, 0);
    float4 p = pv[i], t = tv[i];
    acc += (double)wl_term(p.x, t.x, q0, q1, q2, q3, q4, q5);
    acc += (double)wl_term(p.y, t.y, q0, q1, q2, q3, q4, q5);
    acc += (double)wl_term(p.z, t.z, q0, q1, q2, q3, q4, q5);
    acc += (double)wl_term(p.w, t.w, q0, q1, q2, q3, q4, q5);
  }
  for (unsigned i = (n4 << 2) + gid; i < n; i += stride)
    acc += (double)wl_term(pred[i], targ[i], q0, q1, q2, q3, q4, q5);

  __shared__ float red[TPB];
  __shared__ float red16[16];
  red[threadIdx.x] = (float)acc;
  __syncthreads();

  if (threadIdx.x < 32) {
    float s = 0.0f;
#pragma unroll
    for (int i = 0; i < TPB / 32; ++i) s += red[threadIdx.x * (TPB / 32) + i];
#if WL_HAVE_WMMA
    // D = ones(16x4) * B(4x16) : every row of D holds column sums.
    // Each lane contributes its value once (second B VGPR zeroed), so
    // D[*][n] = s[n] + s[n+16]  -> 32 lane partials reduced to 16.
    wl_v2f a;
    a[0] = 1.0f;
    a[1] = 1.0f;
    wl_v2f b;
    b[0] = s;
    b[1] = 0.0f;
    wl_v8f cc = {};
    cc = __builtin_amdgcn_wmma_f32_16x16x4_f32(false, a, false, b, (short)0, cc,
                                               false, false);
    if (threadIdx.x < 16) red16[threadIdx.x] = cc[0];
#else
#pragma unroll
    for (int off = 16; off >= 1; off >>= 1) s += __shfl_xor(s, off, 32);
    if (threadIdx.x == 0) {
      red16[0] = s;
      for (int i = 1; i < 16; ++i) red16[i] = 0.0f;
    }
#endif
  }
  __syncthreads();
  if (threadIdx.x == 0) {
    float tot = 0.0f;
#pragma unroll
    for (int i = 0; i < 16; ++i) tot += red16[i];
    partials[blockIdx.x] = tot;
  }
}

__global__ void __launch_bounds__(TPB) wl_final(const float* __restrict__ partials,
                                                unsigned nb, float* __restrict__ out,
                                                unsigned n) {
  __shared__ double sd[TPB];
  double s = 0.0;
  for (unsigned i = threadIdx.x; i < nb; i += TPB) s += (double)partials[i];
  sd[threadIdx.x] = s;
  __syncthreads();
  for (int off = TPB / 2; off > 0; off >>= 1) {
    if ((int)threadIdx.x < off) sd[threadIdx.x] += sd[threadIdx.x + off];
    __syncthreads();
  }
  if (threadIdx.x == 0) out[0] = (float)(sd[0] / (double)n);
}

// ---------------------------------------------------------------------------
extern "C" void kernel_launch(void* const* d_in, const int* in_sizes, int n_in,
                              void* d_out, int out_size, void* d_ws, size_t ws_size,
                              hipStream_t stream) {
  (void)n_in;
  (void)out_size;
  (void)ws_size;
  const float* pred = (const float*)d_in[0];
  const float* targ = (const float*)d_in[1];
  const unsigned N = (unsigned)in_sizes[0];

  unsigned char* ws = (unsigned char*)d_ws;
  unsigned* hist1 = (unsigned*)(ws + OFF_HIST1);
  unsigned* hist2 = (unsigned*)(ws + OFF_HIST2);
  unsigned* hist3 = (unsigned*)(ws + OFF_HIST3);
  unsigned* sel1b = (unsigned*)(ws + OFF_SEL1B);
  unsigned* sel1r = (unsigned*)(ws + OFF_SEL1R);
  unsigned* sel2b = (unsigned*)(ws + OFF_SEL2B);
  unsigned* sel2r = (unsigned*)(ws + OFF_SEL2R);
  float* vals = (float*)(ws + OFF_VALS);
  float* quant = (float*)(ws + OFF_QUANT);
  float* partials = (float*)(ws + OFF_PART);

  // zero histograms/selection state every call (graph-capturable memset node)
  (void)hipMemsetAsync(d_ws, 0, ZERO_BYTES, stream);

  // Replicate jnp.quantile's float32 index arithmetic: pos = q * f32(n-1).
  RankInfo ri;
  ri.ranks[0] = 0u;
  ri.ranks[NRANKS - 1] = N - 1u;
  const float probs[4] = {0.2f, 0.4f, 0.6f, 0.8f};
  const float nm1f = (float)(N - 1u);
  for (int j = 1; j <= 4; ++j) {
    float pos = probs[j - 1] * nm1f;
    float fl = floorf(pos);
    unsigned lo = (unsigned)fl;
    if (lo > N - 1u) lo = N - 1u;
    unsigned hi = lo + 1u;
    if (hi > N - 1u) hi = N - 1u;
    ri.ranks[2 * j - 1] = lo;
    ri.ranks[2 * j] = hi;
    ri.frac[j - 1] = pos - fl;
  }

  dim3 grid(GRID_STREAM), block(TPB);
  wl_hist1<<<grid, block, 0, stream>>>(targ, hist1, N);
  wl_sel1<<<1, 32, 0, stream>>>(hist1, sel1b, sel1r, ri);
  wl_hist2<<<grid, block, 0, stream>>>(targ, sel1b, hist2, N);
  wl_sel2<<<1, 32, 0, stream>>>(hist2, sel1r, sel2b, sel2r);
  wl_hist3<<<grid, block, 0, stream>>>(targ, sel1b, sel2b, hist3, N);
  wl_sel3<<<1, 32, 0, stream>>>(hist3, sel1b, sel2b, sel2r, vals, quant, ri);
  wl_mse<<<grid, block, 0, stream>>>(pred, targ, quant, partials, N);
  wl_final<<<1, TPB, 0, stream>>>(partials, GRID_STREAM, (float*)d_out, N);
}